// SimpleFusedConstrainNetwork_55843164783016
// MI455X (gfx1250) — compile-verified
//
#include <hip/hip_runtime.h>

// CDNA5 / gfx1250, wave32. Fused single-GEMM formulation:
//   s = z @ At            (V_WMMA_F32_16X16X4_F32)
//   sign = s - b ; denom = min(pAt - s, -1e-9)
//   alpha_raw = -sign * rcp(denom)   // >= 0 iff violated, < 0 iff satisfied
//   out = z + max(0, rowmax(alpha_raw)) * (pivot - z)
// Memory-bound target: ~134 MB HBM traffic -> ~5.8 us @ 23.3 TB/s.

typedef __attribute__((ext_vector_type(2))) float v2f;
typedef __attribute__((ext_vector_type(8))) float v8f;

#define NROWS 262144
#define D 64
#define M 256
#define CLAMPV (-1e-9f)

#define WAVES_PER_WG 8
#define ROWS_PER_WAVE 16
#define ROWS_PER_WG (WAVES_PER_WG * ROWS_PER_WAVE) /* 128 */
#define ZT_STRIDE 66 /* 64 + 2 pad floats: rotates LDS banks by 2 per row */

__global__ __launch_bounds__(256) void constrain_wmma_kernel(
    const float* __restrict__ z, const float* __restrict__ pivot,
    const float* __restrict__ At, const float* __restrict__ b,
    float* __restrict__ out)
{
    __shared__ float pivsh[D];
    __shared__ float bsh[M];
    __shared__ float pAtsh[M];
    __shared__ float zt[WAVES_PER_WG][ROWS_PER_WAVE][ZT_STRIDE];
    __shared__ float amax[WAVES_PER_WG][ROWS_PER_WAVE];

    const int tid  = threadIdx.x;
    const int wave = tid >> 5;
    const int lane = tid & 31;

    // ---- stage pivot & b, then precompute pAt = pivot @ At (tiny GEMV) ----
    if (tid < D) pivsh[tid] = pivot[tid];
    if (tid < M) bsh[tid]   = b[tid];
    __syncthreads();
    {
        float acc = 0.f; // one m-column per thread, coalesced At reads
        #pragma unroll 8
        for (int d = 0; d < D; ++d) acc += pivsh[d] * At[d * M + tid];
        pAtsh[tid] = acc;
    }
    __syncthreads();

    const int row0 = blockIdx.x * ROWS_PER_WG + wave * ROWS_PER_WAVE;

    // ---- stage this wave's 16x64 z tile into LDS (coalesced 256B per row) ----
    #pragma unroll 4
    for (int r = 0; r < ROWS_PER_WAVE; ++r) {
        v2f v = *(const v2f*)(z + (size_t)(row0 + r) * D + lane * 2);
        *(v2f*)(&zt[wave][r][lane * 2]) = v;
    }

    // ---- A fragments: 16x4 f32 tiles, whole K=64 resident in 32 VGPRs ----
    // A layout (16x4 f32): lanes 0-15 -> M=lane, K={4k,4k+1}; lanes 16-31 -> M=lane-16, K={4k+2,4k+3}
    const int lrow = lane & 15;
    const int koff = (lane >> 4) << 1; // 0 or 2
    v2f afrag[16];
    #pragma unroll
    for (int ks = 0; ks < 16; ++ks)
        afrag[ks] = *(const v2f*)(&zt[wave][lrow][4 * ks + koff]);

    // ---- main loop over 16 column tiles of M=256 ----
    float rmax[8];
    #pragma unroll
    for (int v = 0; v < 8; ++v) rmax[v] = 0.f; // violated alphas are >= 0; satisfied give < 0

    const int col = lrow; // per-lane constraint column within tile
    #pragma unroll 1
    for (int mt = 0; mt < 16; ++mt) {
        const int mcol = mt * 16 + col;
        // B layout (4x16 f32): lanes 0-15 -> K={4k,4k+1}; lanes 16-31 -> K={4k+2,4k+3}; N=lane&15
        const float* Bp = At + (size_t)koff * M + mcol;

        v8f c0 = {0.f, 0.f, 0.f, 0.f, 0.f, 0.f, 0.f, 0.f};
        v8f c1 = {0.f, 0.f, 0.f, 0.f, 0.f, 0.f, 0.f, 0.f};
        #pragma unroll
        for (int ks = 0; ks < 16; ks += 2) {
            v2f b0, b1;
            b0.x = Bp[(4 * ks + 0) * M];
            b0.y = Bp[(4 * ks + 1) * M];
            b1.x = Bp[(4 * ks + 4) * M];
            b1.y = Bp[(4 * ks + 5) * M];
            // (neg_a, A, neg_b, B, c_mod, C, reuse_a, reuse_b)
            c0 = __builtin_amdgcn_wmma_f32_16x16x4_f32(
                false, afrag[ks],     false, b0, (short)0, c0, false, false);
            c1 = __builtin_amdgcn_wmma_f32_16x16x4_f32(
                false, afrag[ks + 1], false, b1, (short)0, c1, false, false);
        }

        // fused branchless epilogue for this 16x16 tile of s = z@At
        const float bcol = bsh[mcol];
        const float pcol = pAtsh[mcol];
        v8f cc = c0 + c1;
        #pragma unroll
        for (int v = 0; v < 8; ++v) {
            float s   = cc[v];
            float sgn = s - bcol;                   // sign_arg
            float den = fminf(pcol - s, CLAMPV);    // clamped ray@At, always < 0
            float r   = __builtin_amdgcn_rcpf(den); // v_rcp_f32 (den <= -1e-9, finite)
            // -sgn*r >= 0 iff sgn >= 0 (violated); < 0 otherwise -> max vs 0 self-masks
            rmax[v]   = fmaxf(rmax[v], -sgn * r);
        }
    }

    // ---- row-max reduction across the 16-lane groups (C layout: VGPR v = rows v / v+8) ----
    #pragma unroll
    for (int v = 0; v < 8; ++v) {
        float m = rmax[v];
        m = fmaxf(m, __shfl_xor(m, 1, 32));
        m = fmaxf(m, __shfl_xor(m, 2, 32));
        m = fmaxf(m, __shfl_xor(m, 4, 32));
        m = fmaxf(m, __shfl_xor(m, 8, 32));
        rmax[v] = m;
    }
    if (col == 0) { // lanes 0 (rows 0..7) and 16 (rows 8..15)
        const int rbase = (lane >> 4) * 8;
        #pragma unroll
        for (int v = 0; v < 8; ++v) amax[wave][rbase + v] = rmax[v];
    }
    // wave-private LDS; DScnt ordering within the wave covers the RAW below

    // ---- output: out = z + amax * (pivot - z), coalesced float2 rows ----
    v2f piv2 = *(const v2f*)(&pivsh[lane * 2]);
    #pragma unroll 4
    for (int r = 0; r < ROWS_PER_WAVE; ++r) {
        float a = amax[wave][r];
        v2f zv  = *(const v2f*)(&zt[wave][r][lane * 2]);
        v2f o;
        o.x = zv.x + a * (piv2.x - zv.x);
        o.y = zv.y + a * (piv2.y - zv.y);
        *(v2f*)(out + (size_t)(row0 + r) * D + lane * 2) = o;
    }
}

extern "C" void kernel_launch(void* const* d_in, const int* in_sizes, int n_in,
                              void* d_out, int out_size, void* d_ws, size_t ws_size,
                              hipStream_t stream) {
    const float* zp  = (const float*)d_in[0]; // [N, D]
    const float* pv  = (const float*)d_in[1]; // [D]
    const float* At  = (const float*)d_in[2]; // [D, M]
    const float* bp  = (const float*)d_in[3]; // [M]
    float* outp = (float*)d_out;              // [N, D]

    dim3 grid(NROWS / ROWS_PER_WG); // 2048 blocks x 256 threads (8 wave32)
    constrain_wmma_kernel<<<grid, dim3(256), 0, stream>>>(zp, pv, At, bp, outp);
}